// _ConvBlock6_43018392436869
// MI455X (gfx1250) — compile-verified
//
#include <hip/hip_runtime.h>

typedef float v2f __attribute__((ext_vector_type(2)));
typedef float v8f __attribute__((ext_vector_type(8)));

// ---------------------------------------------------------------- zero fill
__global__ void zero_kernel(float4* __restrict__ p, long long n4) {
    long long i = (long long)blockIdx.x * blockDim.x + threadIdx.x;
    long long stride = (long long)gridDim.x * blockDim.x;
    float4 z = make_float4(0.f, 0.f, 0.f, 0.f);
    for (; i < n4; i += stride) p[i] = z;
}

// --------------------------------------------- edge gather * w -> scatter-add
// One wave32 per edge: 64 channels = 2 coalesced 128B halves per wave.
__global__ void edge_scatter_kernel(const float* __restrict__ x,
                                    const long long* __restrict__ src,
                                    const long long* __restrict__ dst,
                                    const float* __restrict__ w,
                                    float* __restrict__ agg, int E) {
    long long gid = (long long)blockIdx.x * blockDim.x + threadIdx.x;
    int e = (int)(gid >> 5);
    int lane = (int)(gid & 31);
    if (e >= E) return;
    long long s = src[e];
    long long d = dst[e];
    float ea = w[e];
    const float* xr = x + s * 64;
    float* ar = agg + d * 64;
    atomicAdd(ar + lane,      ea * xr[lane]);
    atomicAdd(ar + 32 + lane, ea * xr[32 + lane]);
}

// ------------------------------------------------- h = agg @ W + b, in place
// One wave per 16-row stripe; 4 accumulator tiles cover all 64 output cols.
// V_WMMA_F32_16X16X4_F32, K looped 0..60 step 4, fully unrolled (64 WMMA/wave).
__global__ void gemm_bias_kernel(float* __restrict__ hb,          // in: agg, out: h
                                 const float* __restrict__ Wm,    // 64x64 row-major
                                 const float* __restrict__ bias,  // 64
                                 int M) {
    long long gid = (long long)blockIdx.x * blockDim.x + threadIdx.x;
    int wave = (int)(gid >> 5);
    int lane = threadIdx.x & 31;
    int row0 = wave * 16;
    if (row0 >= M) return;

    int l15   = lane & 15;
    int hi    = lane >> 4;   // 0 for lanes 0-15, 1 for 16-31
    int khalf = hi * 2;      // A/B K sub-pair per half-wave

    int arow = row0 + l15;
    if (arow >= M) arow = M - 1;                 // M is a multiple of 16 here
    const float* aptr = hb + (long long)arow * 64 + khalf;

    v8f acc0 = {}, acc1 = {}, acc2 = {}, acc3 = {};
#pragma unroll
    for (int kk = 0; kk < 64; kk += 4) {
        // A 16x4: lanes 0-15 -> K=kk,kk+1 ; lanes 16-31 -> K=kk+2,kk+3
        v2f a;
        a.x = aptr[kk];
        a.y = aptr[kk + 1];
        // B 4x16 per col-tile: VGPR0 = row (kk+khalf), VGPR1 = row (kk+khalf+1)
        const float* bp = Wm + (long long)(kk + khalf) * 64 + l15;
        v2f b0; b0.x = bp[0];  b0.y = bp[64];
        v2f b1; b1.x = bp[16]; b1.y = bp[80];
        v2f b2; b2.x = bp[32]; b2.y = bp[96];
        v2f b3; b3.x = bp[48]; b3.y = bp[112];
        acc0 = __builtin_amdgcn_wmma_f32_16x16x4_f32(false, a, false, b0, (short)0, acc0, false, false);
        acc1 = __builtin_amdgcn_wmma_f32_16x16x4_f32(false, a, false, b1, (short)0, acc1, false, false);
        acc2 = __builtin_amdgcn_wmma_f32_16x16x4_f32(false, a, false, b2, (short)0, acc2, false, false);
        acc3 = __builtin_amdgcn_wmma_f32_16x16x4_f32(false, a, false, b3, (short)0, acc3, false, false);
    }

    float bv0 = bias[l15];
    float bv1 = bias[16 + l15];
    float bv2 = bias[32 + l15];
    float bv3 = bias[48 + l15];
    // C/D layout: VGPR v -> M = v + hi*8, N = l15 (+tile*16)
#pragma unroll
    for (int v = 0; v < 8; ++v) {
        int m = row0 + v + hi * 8;
        if (m < M) {
            float* orow = hb + (long long)m * 64 + l15;
            orow[0]  = acc0[v] + bv0;
            orow[16] = acc1[v] + bv1;
            orow[32] = acc2[v] + bv2;
            orow[48] = acc3[v] + bv3;
        }
    }
}

// ------------------------------------------------------- out[fwd[i]] = h[i]
__global__ void inject_kernel(const float* __restrict__ h,
                              const long long* __restrict__ fwd,
                              float* __restrict__ out, int Nc) {
    long long gid = (long long)blockIdx.x * blockDim.x + threadIdx.x;
    int i = (int)(gid >> 5);
    int lane = (int)(gid & 31);
    if (i >= Nc) return;
    long long f = fwd[i];
    const float* hr = h + (long long)i * 64;
    float* orow = out + f * 64;
    orow[lane]      = hr[lane];
    orow[32 + lane] = hr[32 + lane];
}

extern "C" void kernel_launch(void* const* d_in, const int* in_sizes, int n_in,
                              void* d_out, int out_size, void* d_ws, size_t ws_size,
                              hipStream_t stream) {
    const float*     x     = (const float*)d_in[0];
    const float*     Wm    = (const float*)d_in[1];
    const float*     bias  = (const float*)d_in[2];
    const long long* eidx  = (const long long*)d_in[3];   // int64 [2,E]
    const float*     eattr = (const float*)d_in[4];
    const long long* psrc  = (const long long*)d_in[5];   // int64 [E]
    const long long* pdst  = (const long long*)d_in[6];   // int64 [E]
    const float*     pattr = (const float*)d_in[7];
    const long long* fwd   = (const long long*)d_in[8];   // int64 [N_coarse]
    float* out = (float*)d_out;

    const int E  = in_sizes[4];
    const int Nc = in_sizes[0] / 64;
    const int Nf = out_size / 64;

    float* agg = (float*)d_ws;   // Nc*64 floats; reused in-place as h

    const int threads = 256;

    // 1) zero coarse accumulator
    {
        long long n4 = (long long)Nc * 16;
        int blocks = (int)((n4 + threads - 1) / threads);
        zero_kernel<<<blocks, threads, 0, stream>>>((float4*)agg, n4);
    }
    // 2) coarse scatter-add: agg[dst] += edge_attr * x[src]
    {
        long long tot = (long long)E * 32;
        int blocks = (int)((tot + threads - 1) / threads);
        edge_scatter_kernel<<<blocks, threads, 0, stream>>>(x, eidx, eidx + E, eattr, agg, E);
    }
    // 3) h = agg @ W + b (WMMA, in place)
    {
        int waves = (Nc + 15) / 16;
        long long tot = (long long)waves * 32;
        int blocks = (int)((tot + threads - 1) / threads);
        gemm_bias_kernel<<<blocks, threads, 0, stream>>>(agg, Wm, bias, Nc);
    }
    // 4) zero fine output
    {
        long long n4 = (long long)Nf * 16;
        int blocks = (int)((n4 + threads - 1) / threads);
        zero_kernel<<<blocks, threads, 0, stream>>>((float4*)out, n4);
    }
    // 5) fine scatter-add: out[pool_dst] += pool_edge_attr * h[pool_src]
    {
        long long tot = (long long)E * 32;
        int blocks = (int)((tot + threads - 1) / threads);
        edge_scatter_kernel<<<blocks, threads, 0, stream>>>(agg, psrc, pdst, pattr, out, E);
    }
    // 6) direct injection: out[fwd[i]] = h[i]  (after all scatter atomics)
    {
        long long tot = (long long)Nc * 32;
        int blocks = (int)((tot + threads - 1) / threads);
        inject_kernel<<<blocks, threads, 0, stream>>>(agg, fwd, out, Nc);
    }
}